// LengthRegulator_13915694039148
// MI455X (gfx1250) — compile-verified
//
#include <hip/hip_runtime.h>
#include <hip/hip_bf16.h>

// LengthRegulator for MI455X (gfx1250, wave32).
// Pure data-movement op: 128 MiB NT streamed stores + L2-resident gather.
// No matmul content -> WMMA intentionally not used; optimize the memory path.

#define LR_B      32
#define LR_T      512
#define LR_D      256
#define LR_MAXLEN 4096           // T * MAX_DUR
#define LR_D4     (LR_D / 4)     // 64 float4 per row
#define LR_ROWS_PER_BLK 16
#define LR_WAVE   32             // gfx1250 is wave32-only

typedef float v4f __attribute__((ext_vector_type(4)));
typedef int   v4i __attribute__((ext_vector_type(4)));

// ---------------------------------------------------------------------------
// Kernel 1: per-batch inclusive scan of durations.
// ONE wave32 per batch, zero barriers, zero LDS:
//   lane l owns 16 contiguous durations (4 x b128 loads), serial register
//   prefix, then 5-step __shfl_up wave scan of lane totals (lane permutes).
// Writes cum[b][t] (4 x b128 stores) and mel_len[b] (as float) to d_out tail.
// ---------------------------------------------------------------------------
__global__ __launch_bounds__(LR_WAVE) void lr_scan_kernel(
    const int* __restrict__ dur, int* __restrict__ cum,
    float* __restrict__ mel_len) {
  const int b    = blockIdx.x;
  const int lane = threadIdx.x;          // 0..31
  const int base = b * LR_T + lane * 16; // 64B-aligned

  // Load 16 durations as 4 x int4, clamp to >= 0 (jnp.maximum(duration, 0)).
  int v[16];
  const v4i* dp = (const v4i*)(dur + base);
  #pragma unroll
  for (int q = 0; q < 4; ++q) {
    v4i d = dp[q];
    v[q * 4 + 0] = d.x > 0 ? d.x : 0;
    v[q * 4 + 1] = d.y > 0 ? d.y : 0;
    v[q * 4 + 2] = d.z > 0 ? d.z : 0;
    v[q * 4 + 3] = d.w > 0 ? d.w : 0;
  }

  // Serial inclusive prefix within the lane's 16 elements.
  #pragma unroll
  for (int q = 1; q < 16; ++q) v[q] += v[q - 1];
  int total = v[15];

  // Wave-level inclusive scan of lane totals (5 steps, no barriers).
  #pragma unroll
  for (int off = 1; off < LR_WAVE; off <<= 1) {
    const int n = __shfl_up(total, off, LR_WAVE);
    if (lane >= off) total += n;
  }
  const int ofs = total - v[15];   // exclusive offset for this lane

  // Store cum as 4 x int4.
  v4i* cp = (v4i*)(cum + base);
  #pragma unroll
  for (int q = 0; q < 4; ++q) {
    v4i c;
    c.x = v[q * 4 + 0] + ofs;
    c.y = v[q * 4 + 1] + ofs;
    c.z = v[q * 4 + 2] + ofs;
    c.w = v[q * 4 + 3] + ofs;
    cp[q] = c;
  }

  if (lane == LR_WAVE - 1) mel_len[b] = (float)total;  // inclusive total
}

// ---------------------------------------------------------------------------
// Kernel 2: expansion gather.
// Grid: (LR_MAXLEN / 16, B). Block: 256 threads = 8 waves.
// Block stages the 2KB cum row in LDS; 16 threads each do ONE 9-step
// power-of-two search and publish srcRow (or -1 if p >= mel_len) to LDS.
// Then each 64-thread group streams one row: prefetch (global_prefetch_b8)
// -> b128 gather load -> non-temporal b128 store (preserves L2 for x).
// ---------------------------------------------------------------------------
__global__ __launch_bounds__(256) void lr_expand_kernel(
    const v4f* __restrict__ x, const int* __restrict__ cum,
    v4f* __restrict__ out) {
  __shared__ int s_cum[LR_T];
  __shared__ int s_row[LR_ROWS_PER_BLK];   // gather source row, -1 = masked

  const int b   = blockIdx.y;
  const int tid = threadIdx.x;

  // Stage cum row (coalesced b32 loads).
  s_cum[tid]       = cum[b * LR_T + tid];
  s_cum[tid + 256] = cum[b * LR_T + tid + 256];
  __syncthreads();

  const int row0 = blockIdx.x * LR_ROWS_PER_BLK;

  if (tid < LR_ROWS_PER_BLK) {
    const int p   = row0 + tid;
    const int mel = s_cum[LR_T - 1];
    // count = |{t : cum[t] <= p}| via branchless 9-bit greedy search.
    int lo = 0;
    #pragma unroll
    for (int step = 256; step >= 1; step >>= 1) {
      const int nxt = lo + step;
      if (nxt <= LR_T && s_cum[nxt - 1] <= p) lo = nxt;
    }
    // min(count, T-1): count 511 vs 512 both clamp to 511; p >= mel masked.
    const int idx = lo < (LR_T - 1) ? lo : (LR_T - 1);
    s_row[tid] = (p < mel) ? idx : -1;
  }
  __syncthreads();

  const int lane_d = tid & (LR_D4 - 1);   // float4 slot within the row
  const int sub    = tid >> 6;            // which of 4 concurrent rows

  const v4f* __restrict__ xb = x   + (size_t)b * LR_T      * LR_D4;
  v4f* __restrict__       ob = out + (size_t)b * LR_MAXLEN * LR_D4;

  int rows[4];
  #pragma unroll
  for (int i = 0; i < 4; ++i) {
    rows[i] = s_row[i * 4 + sub];
    if (rows[i] >= 0) {
      __builtin_prefetch(&xb[(size_t)rows[i] * LR_D4 + lane_d], 0, 3);
    }
  }

  #pragma unroll
  for (int i = 0; i < 4; ++i) {
    const int p = row0 + i * 4 + sub;
    v4f v = (v4f)(0.0f);
    if (rows[i] >= 0) {
      v = xb[(size_t)rows[i] * LR_D4 + lane_d];
    }
    __builtin_nontemporal_store(v, &ob[(size_t)p * LR_D4 + lane_d]);
  }
}

// ---------------------------------------------------------------------------
// Launch: inputs are (x: f32[32,512,256], duration: i32[32,512]).
// d_out = out[32,4096,256] (33,554,432 f32) ++ mel_len[32] (f32).
// d_ws holds cum[32][512] (64 KiB of int).
// ---------------------------------------------------------------------------
extern "C" void kernel_launch(void* const* d_in, const int* in_sizes, int n_in,
                              void* d_out, int out_size, void* d_ws, size_t ws_size,
                              hipStream_t stream) {
  (void)in_sizes; (void)n_in; (void)out_size; (void)ws_size;

  const v4f* x   = (const v4f*)d_in[0];
  const int* dur = (const int*)d_in[1];
  float*     out = (float*)d_out;
  int*       cum = (int*)d_ws;
  float*     mel = out + (size_t)LR_B * LR_MAXLEN * LR_D;

  lr_scan_kernel<<<dim3(LR_B), dim3(LR_WAVE), 0, stream>>>(dur, cum, mel);

  lr_expand_kernel<<<dim3(LR_MAXLEN / LR_ROWS_PER_BLK, LR_B), dim3(256), 0, stream>>>(
      x, cum, (v4f*)out);
}